// MaxPool2d_malo_21148418966240
// MI455X (gfx1250) — compile-verified
//
#include <hip/hip_runtime.h>

// MaxPool2d k=2, s=2, valid padding over (N=32, C=64, H=224, W=224) fp32.
// Memory-bound streaming kernel: ~514 MB traffic -> ~22 us floor at 23.3 TB/s.
// Strategy: per-thread 2x2 output patch via four NT float4 loads + two NT
// float2 stores, lane-contiguous for full wave32 coalescing.

typedef float f32x4 __attribute__((ext_vector_type(4)));
typedef float f32x2 __attribute__((ext_vector_type(2)));

namespace {
constexpr int N  = 32;
constexpr int C  = 64;
constexpr int IH = 224;
constexpr int IW = 224;
constexpr int OH = IH / 2;           // 112
constexpr int OW = IW / 2;           // 112
constexpr int PX = OW / 2;           // 56 column-pairs per output row
constexpr int PY = OH / 2;           // 56 row-pairs per output image
constexpr int TOTAL_THREADS = N * C * PY * PX;      // 6,422,528
constexpr int IN_ROW_F4  = IW / 4;   // 56 float4 per input row
constexpr int OUT_ROW_F2 = OW / 2;   // 56 float2 per output row
}  // namespace

__global__ __launch_bounds__(256) void maxpool2x2_nt_kernel(
    const float* __restrict__ in, float* __restrict__ out) {
  unsigned tid = blockIdx.x * 256u + threadIdx.x;
  if (tid >= (unsigned)TOTAL_THREADS) return;

  // Decompose: tid -> (nc, row-pair, col-pair). All divisors are
  // compile-time constants -> mul/shift, no integer divide.
  unsigned px  = tid % PX;           // which pair of output columns
  unsigned t   = tid / PX;
  unsigned pyy = t % PY;             // which pair of output rows
  unsigned nc  = t / PY;             // fused batch*channel index

  unsigned ih0 = pyy * 4u;           // first of 4 input rows consumed
  unsigned oh0 = pyy * 2u;           // first of 2 output rows produced

  // 16B-aligned: element offset is a multiple of 4.
  const f32x4* ip = (const f32x4*)(in + (size_t)nc * (IH * IW) +
                                   (size_t)ih0 * IW + (size_t)px * 4u);

  // Four input rows, same 4-column window. Lane-adjacent threads hit
  // adjacent 16B chunks -> each global_load_b128 is fully coalesced.
  f32x4 r0 = __builtin_nontemporal_load(ip + 0 * IN_ROW_F4);
  f32x4 r1 = __builtin_nontemporal_load(ip + 1 * IN_ROW_F4);
  f32x4 r2 = __builtin_nontemporal_load(ip + 2 * IN_ROW_F4);
  f32x4 r3 = __builtin_nontemporal_load(ip + 3 * IN_ROW_F4);

  // 2x2 windows: rows (r0,r1) -> output row oh0; rows (r2,r3) -> oh0+1.
  f32x2 o0, o1;
  o0.x = fmaxf(fmaxf(r0.x, r0.y), fmaxf(r1.x, r1.y));
  o0.y = fmaxf(fmaxf(r0.z, r0.w), fmaxf(r1.z, r1.w));
  o1.x = fmaxf(fmaxf(r2.x, r2.y), fmaxf(r3.x, r3.y));
  o1.y = fmaxf(fmaxf(r2.z, r2.w), fmaxf(r3.z, r3.w));

  // 8B-aligned float2 stores, lane-contiguous -> coalesced global_store_b64.
  f32x2* op = (f32x2*)(out + (size_t)nc * (OH * OW) +
                       (size_t)oh0 * OW + (size_t)px * 2u);
  __builtin_nontemporal_store(o0, op);
  __builtin_nontemporal_store(o1, op + OUT_ROW_F2);
}

extern "C" void kernel_launch(void* const* d_in, const int* in_sizes, int n_in,
                              void* d_out, int out_size, void* d_ws, size_t ws_size,
                              hipStream_t stream) {
  (void)in_sizes; (void)n_in; (void)out_size; (void)d_ws; (void)ws_size;
  const float* in = (const float*)d_in[0];
  float* out = (float*)d_out;

  constexpr int kBlock = 256;
  constexpr int kGrid = (TOTAL_THREADS + kBlock - 1) / kBlock;  // 25088 blocks
  maxpool2x2_nt_kernel<<<kGrid, kBlock, 0, stream>>>(in, out);
}